// EdgeDetection_15865609191651
// MI455X (gfx1250) — compile-verified
//
#include <hip/hip_runtime.h>

// Edge detection (gray -> gauss3x3 -> scharr -> |grad|), fused single pass.
// Memory-bound (~1.7 flop/byte): optimized for the MI455X data-movement path:
// async global->LDS tile loads and async LDS->global broadcast stores (NT).
// Block = 64x4 threads; all index math is shift/add (no integer div/mod).

#define TX 64
#define TY 16
#define GW 68   // TX + 4 (2-pixel halo for blur+scharr)
#define GH 20   // TY + 4
#define BW 66   // TX + 2 (1-pixel halo for scharr)
#define BH 18   // TY + 2
#define IMH 1024
#define IMW 1024

__device__ __forceinline__ int reflect101(int i, int n) {
    if (i < 0) i = -i;
    if (i >= n) i = 2 * n - 2 - i;
    return i;
}

__global__ __launch_bounds__(256)
void EdgeDetection_kernel(const float* __restrict__ x, float* __restrict__ out) {
    __shared__ float chan[3 * GH * GW];   // raw channel tiles (reflect-padded)
    __shared__ float gray[GH * GW];       // grayscale tile
    __shared__ float blur[BH * BW];       // gaussian-blurred tile
    __shared__ float magt[TY * TX];       // gradient magnitude tile

    const int tx  = threadIdx.x;          // 0..63
    const int ty  = threadIdx.y;          // 0..3
    const int tx0 = blockIdx.x * TX;
    const int ty0 = blockIdx.y * TY;
    const int b   = blockIdx.z;

    const unsigned long long inBase  = (unsigned long long)x;
    const unsigned long long outBase = (unsigned long long)out;
    // Low 32 bits of the flat address of a shared object == LDS byte offset.
    const unsigned chanLds = (unsigned)(size_t)(void*)&chan[0];
    const unsigned magLds  = (unsigned)(size_t)(void*)&magt[0];

    // ---- Stage 1: async DMA of 3 channel tiles (reflect-101 halo) into LDS
    // gy*IMW and plane offsets are pure shifts (IMW = IMH = 1024).
    #pragma unroll
    for (int c = 0; c < 3; ++c) {
        const int planeElt = ((b * 3 + c) << 20);           // (b*3+c)*IMH*IMW
        for (int ly = ty; ly < GH; ly += 4) {
            const int gy = reflect101(ty0 - 2 + ly, IMH);
            const int rowElt = planeElt + (gy << 10);
            #pragma unroll
            for (int lxb = 0; lxb < GW; lxb += 64) {
                const int lx = tx + lxb;
                if (lx < GW) {
                    const int gx = reflect101(tx0 - 2 + lx, IMW);
                    const unsigned voff   = (unsigned)((rowElt + gx) << 2);
                    const unsigned ldsOff = chanLds +
                        (unsigned)(((c * GH + ly) * GW + lx) << 2);
                    asm volatile("global_load_async_to_lds_b32 %0, %1, %2"
                                 :: "v"(ldsOff), "v"(voff), "s"(inBase) : "memory");
                }
            }
        }
    }
    asm volatile("s_wait_asynccnt 0" ::: "memory");
    __syncthreads();

    // ---- Stage 2: grayscale (cv2 RGB2GRAY weights)
    for (int ly = ty; ly < GH; ly += 4) {
        #pragma unroll
        for (int lxb = 0; lxb < GW; lxb += 64) {
            const int lx = tx + lxb;
            if (lx < GW) {
                const int i = ly * GW + lx;
                gray[i] = 0.299f * chan[i]
                        + 0.587f * chan[GH * GW + i]
                        + 0.114f * chan[2 * GH * GW + i];
            }
        }
    }
    __syncthreads();

    // ---- Stage 3: separable 3x3 Gaussian, sigma = 0.8 (cv2 auto-sigma, k=3)
    // 1D taps [k0,k1,k0]; symmetric kernel => computing the blur at reflect-
    // padded halo positions equals reflect-padding the blurred image.
    const float k0 = 0.23899420f;
    const float k1 = 0.52201160f;
    for (int by = ty; by < BH; by += 4) {
        #pragma unroll
        for (int bxb = 0; bxb < BW; bxb += 64) {
            const int bx = tx + bxb;
            if (bx < BW) {
                const float* g = &gray[by * GW + bx];   // 3x3 window top-left
                float r0 = k0 * g[0]      + k1 * g[1]          + k0 * g[2];
                float r1 = k0 * g[GW]     + k1 * g[GW + 1]     + k0 * g[GW + 2];
                float r2 = k0 * g[2 * GW] + k1 * g[2 * GW + 1] + k0 * g[2 * GW + 2];
                blur[by * BW + bx] = k0 * r0 + k1 * r1 + k0 * r2;
            }
        }
    }
    __syncthreads();

    // ---- Stage 4: Scharr X/Y + L2 magnitude (ox == threadIdx.x exactly)
    #pragma unroll
    for (int oy = ty; oy < TY; oy += 4) {
        const float* bl = &blur[oy * BW + tx];          // 3x3 window top-left
        float tdx = bl[2]          - bl[0];
        float mdx = bl[BW + 2]     - bl[BW];
        float bdx = bl[2 * BW + 2] - bl[2 * BW];
        float sx  = 3.0f * tdx + 10.0f * mdx + 3.0f * bdx;
        float ldy = bl[2 * BW]     - bl[0];
        float cdy = bl[2 * BW + 1] - bl[1];
        float rdy = bl[2 * BW + 2] - bl[2];
        float sy  = 3.0f * ldy + 10.0f * cdy + 3.0f * rdy;
        magt[(oy << 6) + tx] = sqrtf(sx * sx + sy * sy);
    }
    __syncthreads();

    // ---- Stage 5: async DMA broadcast of the magnitude tile to all 3 output
    // channels, non-temporal (written once, never re-read -> keep L2 for input).
    #pragma unroll
    for (int c = 0; c < 3; ++c) {
        const int planeElt = ((b * 3 + c) << 20);
        #pragma unroll
        for (int oy = ty; oy < TY; oy += 4) {
            const unsigned voff   = (unsigned)((planeElt + ((ty0 + oy) << 10)
                                                + tx0 + tx) << 2);
            const unsigned ldsOff = magLds + (unsigned)(((oy << 6) + tx) << 2);
            asm volatile("global_store_async_from_lds_b32 %0, %1, %2 th:TH_STORE_NT"
                         :: "v"(voff), "v"(ldsOff), "s"(outBase) : "memory");
        }
    }
    asm volatile("s_wait_asynccnt 0" ::: "memory");
}

extern "C" void kernel_launch(void* const* d_in, const int* in_sizes, int n_in,
                              void* d_out, int out_size, void* d_ws, size_t ws_size,
                              hipStream_t stream) {
    const float* x = (const float*)d_in[0];
    float* out = (float*)d_out;
    int batch = in_sizes[0] / (3 * IMH * IMW);   // = 8 for the reference shapes
    dim3 grid(IMW / TX, IMH / TY, batch);
    EdgeDetection_kernel<<<grid, dim3(64, 4), 0, stream>>>(x, out);
}